// HeteroGNN_65352222376662
// MI455X (gfx1250) — compile-verified
//
#include <hip/hip_runtime.h>

typedef float v2f __attribute__((ext_vector_type(2)));
typedef float v4f __attribute__((ext_vector_type(4)));
typedef float v8f __attribute__((ext_vector_type(8)));

#define GK 128   // inner dim of every big GEMM (F == H == 128)
#define TM 64    // rows per block (main GEMM)
#define TMH 128  // rows per block (head GEMM)
#define KC 32    // K chunk staged in LDS

// ---------------------------------------------------------------- utilities
__global__ void k_zero(float* __restrict__ p, long n) {
  long i = (long)blockIdx.x * blockDim.x + threadIdx.x;
  long st = (long)gridDim.x * blockDim.x;
  for (; i < n; i += st) p[i] = 0.0f;
}

__global__ void k_count(const int* __restrict__ dst, float* __restrict__ cnt, int E) {
  int t = blockIdx.x * blockDim.x + threadIdx.x;
  if (t < E) atomicAdd(&cnt[dst[t]], 1.0f);
}

__global__ void k_recip(float* __restrict__ c, int n) {
  int t = blockIdx.x * blockDim.x + threadIdx.x;
  if (t < n) c[t] = 1.0f / fmaxf(c[t], 1.0f);
}

// one edge handled by 32 lanes, 4 floats per lane (E*32 threads)
__global__ void k_scatter(const float* __restrict__ xs, const int* __restrict__ src,
                          const int* __restrict__ dst, float* __restrict__ sum, int E) {
  int t = blockIdx.x * blockDim.x + threadIdx.x;
  int e = t >> 5;
  if (e >= E) return;
  int c = (t & 31) << 2;
  int s = src[e], d = dst[e];
  v4f v = *(const v4f*)(xs + (size_t)s * GK + c);
  float* p = sum + (size_t)d * GK + c;
  atomicAdd(p + 0, v.x);
  atomicAdd(p + 1, v.y);
  atomicAdd(p + 2, v.z);
  atomicAdd(p + 3, v.w);
}

// out[i][j] = beta*out[i][j] + alpha * sum_k Wu[i][coloff+k] * Wx[k][j]
// Wu: [128][256] (upd weight), Wx: [128][128], out: [128][128]
__global__ void k_fuse_w(const float* __restrict__ Wu, int coloff,
                         const float* __restrict__ Wx, float alpha, int betaFlag,
                         float* __restrict__ out) {
  int t = blockIdx.x * blockDim.x + threadIdx.x;  // 16384 threads
  int i = t >> 7, j = t & 127;
  float acc = 0.0f;
  for (int k = 0; k < 128; k++)
    acc += Wu[i * 256 + coloff + k] * Wx[k * 128 + j];
  float v = alpha * acc;
  if (betaFlag) v += out[t];
  out[t] = v;
}

// out[i] = beta*out[i] + alpha*( Wu1@bd + Wu2@bs + bu )[i]
__global__ void k_fuse_b(const float* __restrict__ Wu, const float* __restrict__ bd,
                         const float* __restrict__ bs, const float* __restrict__ bu,
                         float alpha, int betaFlag, float* __restrict__ out) {
  int i = threadIdx.x;  // 128 threads, 1 block
  float acc = bu[i];
  for (int k = 0; k < 128; k++)
    acc += Wu[i * 256 + k] * bd[k] + Wu[i * 256 + 128 + k] * bs[k];
  float v = alpha * acc;
  if (betaFlag) v += out[i];
  out[i] = v;
}

// ---------------------------------------------------------------- WMMA GEMM
// Y[M][128] (+)= alpha * (rs ? rs[row] : 1) * (X[M][128] @ W[128][128]^T) + bias
// f32 WMMA 16x16x4. Block: 256 thr (8 waves), 64-row M tile, K chunked by 32.
// Work split is compile-time uniform: wave w -> rows (w&3)*16..+16,
// N tiles (w>>2)*4..+4.  NO runtime conditionals around WMMA (EXEC all-1s).
// A frag (ISA layout): lane m+16h holds A[m][k0+2h .. k0+2h+1] in a v2f.
// B frag:              lane n+16h holds B[k0+2h .. +1][n]      in a v2f,
// so W is staged in LDS pre-paired: Ws[k2][n] = (W[n][2k2], W[n][2k2+1]).
__global__ __launch_bounds__(256)
void k_gemm128(const float* __restrict__ X, const float* __restrict__ W,
               const float* __restrict__ rs, const float* __restrict__ bias,
               float alpha, int betaFlag, float* __restrict__ Y, int M) {
  __shared__ v2f Xs[TM * 18];          // rows padded to 18 v2f -> conflict-free
  __shared__ v2f Ws[(KC / 2) * 128];   // [k2][n], natural stride conflict-free

  const int tid = threadIdx.x;
  const int lane = tid & 31;
  const int wave = tid >> 5;
  const int msub = wave & 3;       // 4 M sub-tiles of 16 rows
  const int ngroup = wave >> 2;    // 2 N groups of 4 tiles (64 cols)
  const int row0 = blockIdx.x * TM;
  const int m = lane & 15;
  const int h = lane >> 4;

  v8f zv = {0.f, 0.f, 0.f, 0.f, 0.f, 0.f, 0.f, 0.f};
  v8f acc0 = zv, acc1 = zv, acc2 = zv, acc3 = zv;

  for (int kk = 0; kk < GK; kk += KC) {
    // stage X tile: 64 rows x 32 cols, float4 coalesced
#pragma unroll
    for (int i = 0; i < 2; i++) {
      int idx = tid + i * 256;           // 0..511
      int r = idx >> 3, c4 = idx & 7;
      int row = row0 + r;
      v4f v = {0.f, 0.f, 0.f, 0.f};
      if (row < M) v = *(const v4f*)(X + (size_t)row * GK + kk + c4 * 4);
      Xs[r * 18 + c4 * 2 + 0] = __builtin_shufflevector(v, v, 0, 1);
      Xs[r * 18 + c4 * 2 + 1] = __builtin_shufflevector(v, v, 2, 3);
    }
    // stage W chunk pre-swizzled: Ws[k2][n] = (W[n][kk+2k2], W[n][kk+2k2+1])
#pragma unroll
    for (int i = 0; i < 8; i++) {
      int idx = tid + i * 256;           // 0..2047
      int n = idx >> 4, k2 = idx & 15;
      Ws[k2 * 128 + n] = *(const v2f*)(W + (size_t)n * GK + kk + k2 * 2);
    }
    __syncthreads();

#pragma unroll
    for (int ks = 0; ks < KC; ks += 4) {
      v2f a = Xs[(msub * 16 + m) * 18 + (ks >> 1) + h];
      const v2f* wrow = &Ws[((ks >> 1) + h) * 128 + ngroup * 64 + m];
      v2f b0 = wrow[0];
      v2f b1 = wrow[16];
      v2f b2 = wrow[32];
      v2f b3 = wrow[48];
      acc0 = __builtin_amdgcn_wmma_f32_16x16x4_f32(false, a, false, b0, (short)0, acc0, false, false);
      acc1 = __builtin_amdgcn_wmma_f32_16x16x4_f32(false, a, false, b1, (short)0, acc1, false, false);
      acc2 = __builtin_amdgcn_wmma_f32_16x16x4_f32(false, a, false, b2, (short)0, acc2, false, false);
      acc3 = __builtin_amdgcn_wmma_f32_16x16x4_f32(false, a, false, b3, (short)0, acc3, false, false);
    }
    __syncthreads();
  }

  // C/D layout: VGPR r, lanes 0-15 -> M=r, lanes 16-31 -> M=r+8; N = lane&15
  v8f accs[4] = {acc0, acc1, acc2, acc3};
#pragma unroll
  for (int t = 0; t < 4; t++) {
    int col = (ngroup * 4 + t) * 16 + m;
#pragma unroll
    for (int r = 0; r < 8; r++) {
      int row = row0 + msub * 16 + h * 8 + r;
      if (row < M) {
        float v = accs[t][r] * alpha;
        if (rs) v *= rs[row];
        if (bias) v += bias[col];
        float* yp = Y + (size_t)row * GK + col;
        if (betaFlag) v += *yp;
        *yp = v;
      }
    }
  }
}

// Head GEMM: Y[M][8] = X[M][128] @ W[8][128]^T + bias.  128-row M tile,
// 8 waves x (16 rows, 1 N tile).  Column mask only in staging & epilogue.
__global__ __launch_bounds__(256)
void k_gemm_head(const float* __restrict__ X, const float* __restrict__ W,
                 const float* __restrict__ bias, float* __restrict__ Y, int M) {
  __shared__ v2f Xs[TMH * 18];
  __shared__ v2f Ws[16 * 16];   // [k2][n], 16 cols (8 valid)

  const int tid = threadIdx.x;
  const int lane = tid & 31;
  const int wave = tid >> 5;   // rows wave*16 .. +16
  const int row0 = blockIdx.x * TMH;
  const int m = lane & 15;
  const int h = lane >> 4;

  v8f acc = {0.f, 0.f, 0.f, 0.f, 0.f, 0.f, 0.f, 0.f};

  for (int kk = 0; kk < GK; kk += KC) {
#pragma unroll
    for (int i = 0; i < 4; i++) {
      int idx = tid + i * 256;           // 0..1023
      int r = idx >> 3, c4 = idx & 7;
      int row = row0 + r;
      v4f v = {0.f, 0.f, 0.f, 0.f};
      if (row < M) v = *(const v4f*)(X + (size_t)row * GK + kk + c4 * 4);
      Xs[r * 18 + c4 * 2 + 0] = __builtin_shufflevector(v, v, 0, 1);
      Xs[r * 18 + c4 * 2 + 1] = __builtin_shufflevector(v, v, 2, 3);
    }
    {
      int n = tid >> 4, k2 = tid & 15;   // 256 threads -> 16x16
      v2f w = {0.f, 0.f};
      if (n < 8) w = *(const v2f*)(W + (size_t)n * GK + kk + k2 * 2);
      Ws[k2 * 16 + n] = w;
    }
    __syncthreads();

#pragma unroll
    for (int ks = 0; ks < KC; ks += 4) {
      v2f a = Xs[(wave * 16 + m) * 18 + (ks >> 1) + h];
      v2f b = Ws[((ks >> 1) + h) * 16 + m];
      acc = __builtin_amdgcn_wmma_f32_16x16x4_f32(false, a, false, b, (short)0, acc, false, false);
    }
    __syncthreads();
  }

  if (m < 8) {
    float bb = bias[m];
#pragma unroll
    for (int r = 0; r < 8; r++) {
      int row = row0 + wave * 16 + h * 8 + r;
      if (row < M) Y[(size_t)row * 8 + m] = acc[r] + bb;
    }
  }
}

// ---------------------------------------------------------------- BatchNorm
// per-column: s = g*rsqrt(var+1), t = b - mean*s  (biased var, batch stats)
__global__ __launch_bounds__(256)
void k_bn_stats(const float* __restrict__ X, const float* __restrict__ g,
                const float* __restrict__ b, float* __restrict__ st, int M) {
  int j = blockIdx.x;  // 0..127
  float s1 = 0.f, s2 = 0.f;
  for (int r = threadIdx.x; r < M; r += 256) {
    float x = X[(size_t)r * 128 + j];
    s1 += x;
    s2 += x * x;
  }
  __shared__ float a1[256], a2[256];
  a1[threadIdx.x] = s1;
  a2[threadIdx.x] = s2;
  __syncthreads();
  for (int o = 128; o > 0; o >>= 1) {
    if (threadIdx.x < o) {
      a1[threadIdx.x] += a1[threadIdx.x + o];
      a2[threadIdx.x] += a2[threadIdx.x + o];
    }
    __syncthreads();
  }
  if (threadIdx.x == 0) {
    float mean = a1[0] / M;
    float var = a2[0] / M - mean * mean;
    float sc = g[j] * rsqrtf(var + 1.0f);
    st[j] = sc;
    st[128 + j] = b[j] - mean * sc;
  }
}

__global__ void k_bn_apply(float* __restrict__ X, const float* __restrict__ st, long n) {
  long t = (long)blockIdx.x * blockDim.x + threadIdx.x;
  if (t < n) {
    int col = (int)(t & 127);
    float x = X[t] * st[col] + st[128 + col];
    X[t] = x >= 0.f ? x : 0.01f * x;
  }
}

// ---------------------------------------------------------------- launcher
extern "C" void kernel_launch(void* const* d_in, const int* in_sizes, int n_in,
                              void* d_out, int out_size, void* d_ws, size_t ws_size,
                              hipStream_t stream) {
  // setup_inputs() dict insertion order, recursively flattened:
  // 0 xA, 1 xB, 2 eAB, 3 eBA, 4 eAA,
  // 5.. conv1 {AB,BA,AA} x {dstW,dstb,srcW,srcb,updW,updb}, 23.. conv2 same,
  // 41.. bn1.A.g, bn1.A.b, bn1.B.g, bn1.B.b, 45.. bn2 same,
  // 49.. post.A.W, post.A.b, post.B.W, post.B.b
  const float* xA = (const float*)d_in[0];
  const float* xB = (const float*)d_in[1];
  const int* eAB = (const int*)d_in[2];
  const int* eBA = (const int*)d_in[3];
  const int* eAA = (const int*)d_in[4];
  const int E_AB = in_sizes[2] / 2, E_BA = in_sizes[3] / 2, E_AA = in_sizes[4] / 2;
  const int N = in_sizes[0] / GK;
  const size_t NH = (size_t)N * GK;
  const int IDX_CONV = 5, IDX_BN = 41, IDX_POST = 49;

  float* p = (float*)d_ws;
  auto take = [&](size_t n) { float* r = p; p += n; return r; };
  float* sum = take(NH);
  float* yA = take(NH); float* yB = take(NH);
  float* zA = take(NH); float* zB = take(NH);
  float* rsAB = take(N); float* rsBA = take(N); float* rsAA = take(N);
  float* wdA = take(16384); float* wsBA = take(16384); float* wsAA = take(16384);
  float* wdB = take(16384); float* wsAB = take(16384);
  float* bAv = take(128); float* bBv = take(128);
  float* st = take(256);

  const int gg = (N + TM - 1) / TM;
  const int gh = (N + TMH - 1) / TMH;
  auto cdiv = [](long a, int b) { return (int)((a + b - 1) / b); };

  // per-relation degree scales (edges are layer-invariant: compute once)
  k_zero<<<256, 256, 0, stream>>>(rsAB, N);
  k_zero<<<256, 256, 0, stream>>>(rsBA, N);
  k_zero<<<256, 256, 0, stream>>>(rsAA, N);
  k_count<<<cdiv(E_AB, 256), 256, 0, stream>>>(eAB + E_AB, rsAB, E_AB);
  k_count<<<cdiv(E_BA, 256), 256, 0, stream>>>(eBA + E_BA, rsBA, E_BA);
  k_count<<<cdiv(E_AA, 256), 256, 0, stream>>>(eAA + E_AA, rsAA, E_AA);
  k_recip<<<cdiv(N, 256), 256, 0, stream>>>(rsAB, N);
  k_recip<<<cdiv(N, 256), 256, 0, stream>>>(rsBA, N);
  k_recip<<<cdiv(N, 256), 256, 0, stream>>>(rsAA, N);

  const float* curA = xA;
  const float* curB = xB;
  float* la[2] = {yA, zA};
  float* lb[2] = {yB, zB};

  for (int l = 0; l < 2; l++) {
    // mt: 0=AB, 1=BA, 2=AA ; w: 0 dstW, 1 dstb, 2 srcW, 3 srcb, 4 updW, 5 updb
    auto P = [&](int mt, int w) {
      return (const float*)d_in[IDX_CONV + l * 18 + mt * 6 + w];
    };
    // fused weights/biases (relation-mean 0.5 folded for dst-A pair)
    k_fuse_w<<<64, 256, 0, stream>>>(P(1, 4), 0,   P(1, 0), 0.5f, 0, wdA);
    k_fuse_w<<<64, 256, 0, stream>>>(P(2, 4), 0,   P(2, 0), 0.5f, 1, wdA);
    k_fuse_w<<<64, 256, 0, stream>>>(P(1, 4), 128, P(1, 2), 1.0f, 0, wsBA);
    k_fuse_w<<<64, 256, 0, stream>>>(P(2, 4), 128, P(2, 2), 1.0f, 0, wsAA);
    k_fuse_w<<<64, 256, 0, stream>>>(P(0, 4), 0,   P(0, 0), 1.0f, 0, wdB);
    k_fuse_w<<<64, 256, 0, stream>>>(P(0, 4), 128, P(0, 2), 1.0f, 0, wsAB);
    k_fuse_b<<<1, 128, 0, stream>>>(P(1, 4), P(1, 1), P(1, 3), P(1, 5), 0.5f, 0, bAv);
    k_fuse_b<<<1, 128, 0, stream>>>(P(2, 4), P(2, 1), P(2, 3), P(2, 5), 0.5f, 1, bAv);
    k_fuse_b<<<1, 128, 0, stream>>>(P(0, 4), P(0, 1), P(0, 3), P(0, 5), 1.0f, 0, bBv);

    float* oA = la[l];
    float* oB = lb[l];
    // type-A output: dst path (BA+AA combined) + two src paths
    k_gemm128<<<gg, 256, 0, stream>>>(curA, wdA, nullptr, bAv, 1.0f, 0, oA, N);
    k_zero<<<2048, 256, 0, stream>>>(sum, (long)NH);
    k_scatter<<<cdiv((long)E_BA * 32, 256), 256, 0, stream>>>(curB, eBA, eBA + E_BA, sum, E_BA);
    k_gemm128<<<gg, 256, 0, stream>>>(sum, wsBA, rsBA, nullptr, 0.5f, 1, oA, N);
    k_zero<<<2048, 256, 0, stream>>>(sum, (long)NH);
    k_scatter<<<cdiv((long)E_AA * 32, 256), 256, 0, stream>>>(curA, eAA, eAA + E_AA, sum, E_AA);
    k_gemm128<<<gg, 256, 0, stream>>>(sum, wsAA, rsAA, nullptr, 0.5f, 1, oA, N);
    // type-B output: dst path + one src path
    k_gemm128<<<gg, 256, 0, stream>>>(curB, wdB, nullptr, bBv, 1.0f, 0, oB, N);
    k_zero<<<2048, 256, 0, stream>>>(sum, (long)NH);
    k_scatter<<<cdiv((long)E_AB * 32, 256), 256, 0, stream>>>(curA, eAB, eAB + E_AB, sum, E_AB);
    k_gemm128<<<gg, 256, 0, stream>>>(sum, wsAB, rsAB, nullptr, 1.0f, 1, oB, N);

    // BN (train-mode batch stats, eps=1) + LeakyReLU, in place
    const float* gA = (const float*)d_in[IDX_BN + l * 4 + 0];
    const float* bA_ = (const float*)d_in[IDX_BN + l * 4 + 1];
    const float* gB = (const float*)d_in[IDX_BN + l * 4 + 2];
    const float* bB_ = (const float*)d_in[IDX_BN + l * 4 + 3];
    k_bn_stats<<<128, 256, 0, stream>>>(oA, gA, bA_, st, N);
    k_bn_apply<<<cdiv((long)NH, 256), 256, 0, stream>>>(oA, st, (long)NH);
    k_bn_stats<<<128, 256, 0, stream>>>(oB, gB, bB_, st, N);
    k_bn_apply<<<cdiv((long)NH, 256), 256, 0, stream>>>(oB, st, (long)NH);

    curA = oA;
    curB = oB;
  }

  // post heads 128 -> 8, into stacked [2][N][8] output
  k_gemm_head<<<gh, 256, 0, stream>>>(curA, (const float*)d_in[IDX_POST + 0],
                                      (const float*)d_in[IDX_POST + 1],
                                      (float*)d_out, N);
  k_gemm_head<<<gh, 256, 0, stream>>>(curB, (const float*)d_in[IDX_POST + 2],
                                      (const float*)d_in[IDX_POST + 3],
                                      (float*)d_out + (size_t)N * 8, N);
}